// M2CausalTransformer_34823594836523
// MI455X (gfx1250) — compile-verified
//
#include <hip/hip_runtime.h>
#include <math.h>

typedef float v2f __attribute__((ext_vector_type(2)));
typedef float v4f __attribute__((ext_vector_type(4)));
typedef float v8f __attribute__((ext_vector_type(8)));

#define XD   1024
#define SEQ  4096
#define NBAT 32

static __device__ __forceinline__ v8f wmma_f32_k4(v2f a, v2f b, v8f c) {
  // D = A(16x4, f32) * B(4x16, f32) + C(16x16, f32)
  return __builtin_amdgcn_wmma_f32_16x16x4_f32(
      /*neg_a=*/false, a, /*neg_b=*/false, b,
      /*c_mod=*/(short)0, c, /*reuse_a=*/false, /*reuse_b=*/false);
}

// ---------------------------------------------------------------------------
// scores[b,s] = (x[b,s,:] . x[b,S-1,:]) / 32     (one wave per row-dot)
// x is a 537 MB stream -> non-temporal loads (keep L2 clean for later passes).
// ---------------------------------------------------------------------------
__global__ __launch_bounds__(256) void scores_kernel(const float* __restrict__ x,
                                                     float* __restrict__ scores) {
  __shared__ v4f qs[XD / 4];
  const int b = blockIdx.x;
  const float* xb = x + (long)b * SEQ * XD;
  const float* q  = xb + (long)(SEQ - 1) * XD;
  qs[threadIdx.x] = *(const v4f*)(q + threadIdx.x * 4);
  __syncthreads();
  const int lane = threadIdx.x & 31;
  const int wave = threadIdx.x >> 5;
  for (int rr = 0; rr < 8; ++rr) {
    const int s = blockIdx.y * 64 + wave * 8 + rr;
    const float* xr = xb + (long)s * XD;
    float acc = 0.f;
#pragma unroll
    for (int i = 0; i < 8; ++i) {
      const v4f xv = __builtin_nontemporal_load((const v4f*)(xr + (lane + i * 32) * 4));
      const v4f qv = qs[lane + i * 32];
      acc += xv[0] * qv[0] + xv[1] * qv[1] + xv[2] * qv[2] + xv[3] * qv[3];
    }
#pragma unroll
    for (int off = 16; off; off >>= 1) acc += __shfl_xor(acc, off);
    if (lane == 0) scores[b * SEQ + s] = acc * 0.03125f;  // 1/sqrt(1024)
  }
}

// ---------------------------------------------------------------------------
// softmax over S=4096 per batch (one block per batch)
// ---------------------------------------------------------------------------
__global__ __launch_bounds__(256) void softmax_kernel(const float* __restrict__ scores,
                                                      float* __restrict__ attn) {
  const int b = blockIdx.x;
  const float* sc = scores + b * SEQ;
  const int lane = threadIdx.x & 31, wave = threadIdx.x >> 5;
  float vals[16];
  float m = -INFINITY;
#pragma unroll
  for (int i = 0; i < 16; ++i) {
    vals[i] = sc[threadIdx.x + i * 256];
    m = fmaxf(m, vals[i]);
  }
#pragma unroll
  for (int off = 16; off; off >>= 1) m = fmaxf(m, __shfl_xor(m, off));
  __shared__ float redm[8];
  if (lane == 0) redm[wave] = m;
  __syncthreads();
#pragma unroll
  for (int i = 0; i < 8; ++i) m = fmaxf(m, redm[i]);
  float sum = 0.f;
#pragma unroll
  for (int i = 0; i < 16; ++i) { vals[i] = __expf(vals[i] - m); sum += vals[i]; }
#pragma unroll
  for (int off = 16; off; off >>= 1) sum += __shfl_xor(sum, off);
  __shared__ float reds[8];
  if (lane == 0) reds[wave] = sum;
  __syncthreads();
  float tot = 0.f;
#pragma unroll
  for (int i = 0; i < 8; ++i) tot += reds[i];
  const float inv = 1.f / tot;
#pragma unroll
  for (int i = 0; i < 16; ++i)
    attn[b * SEQ + threadIdx.x + i * 256] = vals[i] * inv;
}

// ---------------------------------------------------------------------------
// v = x @ Wv^T  (fp32 WMMA), fused with sa += attn-weighted row reduction.
// Block: 128 threads = 4 waves (1 wave/SIMD -> launchable at ~330 VGPRs).
// Wave tile: 4 M-tiles x 8 N-tiles (256 acc VGPRs via gfx1250 VGPR-MSB).
// Block covers 64 rows x 512 cols; grid (BS/64, 2).
// Addressing: 2 base pointers + constant immediate offsets (row strides are
// compile-time: 16*XD floats = 64 KB < 2^23 imm) -> only 2 pointer increments
// per k-iter instead of 12.
// x fragments: NT loads (stream); Wv: default RT (stays L2-resident);
// v: NT stores (never re-read on device).
// ---------------------------------------------------------------------------
__global__ __launch_bounds__(128) void v_gemm_sa_kernel(
    const float* __restrict__ x, const float* __restrict__ Wv,
    const float* __restrict__ attn, float* __restrict__ v,
    float* __restrict__ sa) {
  const int lane = threadIdx.x & 31;
  const int wave = threadIdx.x >> 5;       // 0..3
  const int half = lane >> 4;              // K half-select for A/B frags
  const int lr   = lane & 15;              // row/col within 16
  const long m0  = (long)blockIdx.x * 64;
  const int  n0  = blockIdx.y * 512 + wave * 128;

  const float* abase = x  + (m0 + lr) * XD + 2 * half;            // + mt*16*XD + k0
  const float* bbase = Wv + (long)(n0 + lr) * XD + 2 * half;      // + t*16*XD + k0

  const v8f vzero = {0.f, 0.f, 0.f, 0.f, 0.f, 0.f, 0.f, 0.f};
  v8f acc[4][8];
#pragma unroll
  for (int mt = 0; mt < 4; ++mt)
#pragma unroll
    for (int t = 0; t < 8; ++t) acc[mt][t] = vzero;

  for (int k0 = 0; k0 < XD; k0 += 4) {
    v2f a[4];
#pragma unroll
    for (int mt = 0; mt < 4; ++mt)
      a[mt] = __builtin_nontemporal_load((const v2f*)(abase + mt * 16 * XD + k0));
#pragma unroll
    for (int t = 0; t < 8; ++t) {
      const v2f bb = *(const v2f*)(bbase + t * 16 * XD + k0);  // reused across 4 M-tiles
#pragma unroll
      for (int mt = 0; mt < 4; ++mt)
        acc[mt][t] = wmma_f32_k4(a[mt], bb, acc[mt][t]);
    }
  }

  // Epilogue: store v tile + fold attn-weighted partial sums into sa.
  const int batch = (int)(m0 >> 12);          // 4096 rows per batch, no straddle
  const int sbase = (int)(m0 & (SEQ - 1));
  float w[4][8];
#pragma unroll
  for (int mt = 0; mt < 4; ++mt)
#pragma unroll
    for (int r = 0; r < 8; ++r)
      w[mt][r] = attn[batch * SEQ + sbase + mt * 16 + 8 * half + r];

#pragma unroll
  for (int t = 0; t < 8; ++t) {
    float p = 0.f;
#pragma unroll
    for (int mt = 0; mt < 4; ++mt) {
      const v8f c = acc[mt][t];
#pragma unroll
      for (int r = 0; r < 8; ++r) {
        const long row = m0 + mt * 16 + 8 * half + r;  // C layout: VGPR r -> M=r (+8 hi lanes)
        __builtin_nontemporal_store(c[r], v + row * XD + n0 + t * 16 + lr);
        p += w[mt][r] * c[r];
      }
    }
    p += __shfl_xor(p, 16);          // combine lane halves (same N, rows 0-7 vs 8-15)
    if (half == 0)
      unsafeAtomicAdd(&sa[batch * XD + n0 + t * 16 + lr], p);
  }
}

// ---------------------------------------------------------------------------
// h = x[:, -1, :] + sa ; LayerNorm -> hn   (one block per batch)
// ---------------------------------------------------------------------------
__global__ __launch_bounds__(256) void ln_kernel(
    const float* __restrict__ x, const float* __restrict__ sa,
    const float* __restrict__ g, const float* __restrict__ bt,
    float* __restrict__ hn) {
  const int b = blockIdx.x;
  const float* xl  = x + ((long)b * SEQ + (SEQ - 1)) * XD;
  const float* sab = sa + b * XD;
  const int lane = threadIdx.x & 31, wave = threadIdx.x >> 5;
  float h[4];
  float s = 0.f, s2 = 0.f;
#pragma unroll
  for (int i = 0; i < 4; ++i) {
    const int idx = threadIdx.x + i * 256;
    h[i] = xl[idx] + sab[idx];
    s += h[i];
    s2 += h[i] * h[i];
  }
#pragma unroll
  for (int off = 16; off; off >>= 1) { s += __shfl_xor(s, off); s2 += __shfl_xor(s2, off); }
  __shared__ float rs[8], rs2[8];
  if (lane == 0) { rs[wave] = s; rs2[wave] = s2; }
  __syncthreads();
  s = 0.f; s2 = 0.f;
#pragma unroll
  for (int i = 0; i < 8; ++i) { s += rs[i]; s2 += rs2[i]; }
  const float mu  = s * (1.f / XD);
  const float var = s2 * (1.f / XD) - mu * mu;
  const float rin = rsqrtf(var + 1e-5f);
#pragma unroll
  for (int i = 0; i < 4; ++i) {
    const int idx = threadIdx.x + i * 256;
    hn[b * XD + idx] = (h[i] - mu) * rin * g[idx] + bt[idx];
  }
}

// ---------------------------------------------------------------------------
// out[32 x Nvalid] = act(A[32 x K] @ W[Nvalid x K]^T + bias)  (fp32 WMMA)
// Wave: 2 M-tiles x 1 N-tile; block covers 128 output cols.
// W is a one-shot weight stream -> NT loads; A (activations) reused -> RT.
// ---------------------------------------------------------------------------
__global__ __launch_bounds__(256) void fc_wmma_kernel(
    const float* __restrict__ A, const float* __restrict__ W,
    const float* __restrict__ bias, float* __restrict__ out,
    int K, int Nvalid, int relu) {
  const int lane = threadIdx.x & 31;
  const int wave = threadIdx.x >> 5;
  const int half = lane >> 4;
  const int lr   = lane & 15;
  const int n    = blockIdx.x * 128 + wave * 16 + lr;
  const int nc   = min(n, Nvalid - 1);   // clamp row ptr (masked at store)

  const float* a0p = A + lr * K + 2 * half;
  const float* a1p = A + (16 + lr) * K + 2 * half;
  const float* bpp = W + (long)nc * K + 2 * half;

  v8f acc0 = {0.f, 0.f, 0.f, 0.f, 0.f, 0.f, 0.f, 0.f};
  v8f acc1 = acc0;
  for (int k0 = 0; k0 < K; k0 += 4) {
    const v2f a0 = *(const v2f*)(a0p + k0);
    const v2f a1 = *(const v2f*)(a1p + k0);
    const v2f bb = __builtin_nontemporal_load((const v2f*)(bpp + k0));
    acc0 = wmma_f32_k4(a0, bb, acc0);
    acc1 = wmma_f32_k4(a1, bb, acc1);
  }
  if (n < Nvalid) {
    const float bv = bias[n];
#pragma unroll
    for (int r = 0; r < 8; ++r) {
      const int row0 = 8 * half + r;
      float v0 = acc0[r] + bv;
      float v1 = acc1[r] + bv;
      if (relu) { v0 = fmaxf(v0, 0.f); v1 = fmaxf(v1, 0.f); }
      out[(long)row0 * Nvalid + n]        = v0;
      out[(long)(row0 + 16) * Nvalid + n] = v1;
    }
  }
}

// ---------------------------------------------------------------------------
extern "C" void kernel_launch(void* const* d_in, const int* in_sizes, int n_in,
                              void* d_out, int out_size, void* d_ws, size_t ws_size,
                              hipStream_t stream) {
  const float* x    = (const float*)d_in[0];
  const float* Wv   = (const float*)d_in[1];
  const float* ln_g = (const float*)d_in[2];
  const float* ln_b = (const float*)d_in[3];
  const float* W1   = (const float*)d_in[4];
  const float* b1   = (const float*)d_in[5];
  const float* W2   = (const float*)d_in[6];
  const float* b2   = (const float*)d_in[7];
  const float* W3   = (const float*)d_in[8];
  const float* b3   = (const float*)d_in[9];

  // d_out = out (32*1000) | v (32*4096*1024) | attn (32*4096)
  float* out  = (float*)d_out;
  float* v    = out + 32 * 1000;
  float* attn = v + (long)NBAT * SEQ * XD;

  float* ws     = (float*)d_ws;
  float* scores = ws;                    // 32*4096
  float* sa     = scores + NBAT * SEQ;   // 32*1024
  float* hn     = sa + NBAT * XD;        // 32*1024
  float* h1     = hn + NBAT * XD;        // 32*4096
  float* h2     = h1 + NBAT * 4096;      // 32*4096

  hipMemsetAsync(sa, 0, NBAT * XD * sizeof(float), stream);

  scores_kernel <<<dim3(NBAT, SEQ / 64), 256, 0, stream>>>(x, scores);
  softmax_kernel<<<NBAT, 256, 0, stream>>>(scores, attn);
  v_gemm_sa_kernel<<<dim3((NBAT * SEQ) / 64, 2), 128, 0, stream>>>(x, Wv, attn, v, sa);
  ln_kernel     <<<NBAT, 256, 0, stream>>>(x, sa, ln_g, ln_b, hn);
  fc_wmma_kernel<<<4096 / 128, 256, 0, stream>>>(hn, W1, b1, h1, 1024, 4096, 1);
  fc_wmma_kernel<<<4096 / 128, 256, 0, stream>>>(h1, W2, b2, h2, 4096, 4096, 1);
  fc_wmma_kernel<<<1024 / 128, 256, 0, stream>>>(h2, W3, b3, out, 4096, 1000, 0);
}